// MultiHeadReadOutAttention_23854248362063
// MI455X (gfx1250) — compile-verified
//
#include <hip/hip_runtime.h>

typedef float v2f __attribute__((ext_vector_type(2)));
typedef float v8f __attribute__((ext_vector_type(8)));

#define B_   4
#define S_   32
#define H_   8
#define L_   128
#define D_   512
#define DI_  2048
#define N_   (B_ * S_)    // 128 tokens
#define NH_  (N_ * H_)    // 1024 rows for FFN
#define EPS_ 1e-6f
#define INV_TEMP_ 0.04419417382415922f  // 1/sqrt(512)

// ---------------------------------------------------------------------------
// CDNA5 async global->LDS staging (ASYNCcnt path), no VGPR round-trip.
// LDS byte offset = low 32 bits of the generic __shared__ pointer.
// ---------------------------------------------------------------------------
__device__ __forceinline__ void async_load_b128(const void* gptr, void* lptr) {
    asm volatile("global_load_async_to_lds_b128 %0, %1, off"
                 :: "v"((unsigned int)(unsigned long long)lptr),
                    "v"((unsigned long long)gptr)
                 : "memory");
}
__device__ __forceinline__ void wait_async_lds() {
    asm volatile("s_wait_asynccnt 0x0" ::: "memory");
}

// ---------------------------------------------------------------------------
// Kernel 1: attn logits (per-l dot over d) + mask + softmax over l.
// One block per (b,s); 256 threads compute 8x128 logits, 8 waves do softmax.
// ---------------------------------------------------------------------------
__global__ __launch_bounds__(256) void attn_softmax_kernel(
    const float* __restrict__ x, const int* __restrict__ mask,
    const float* __restrict__ w, float* __restrict__ attn_out)
{
    __shared__ float logit[H_][L_];
    const int n = blockIdx.x;
    const int tid = threadIdx.x;
    const float4* xb = (const float4*)(x + (size_t)n * L_ * D_);
    const float4* wb = (const float4*)w;

    #pragma unroll
    for (int it = 0; it < 4; ++it) {
        int idx = tid + it * 256;          // 0..1023
        int h = idx >> 7, l = idx & 127;
        const float4* xr = xb + (size_t)l * (D_ / 4);
        const float4* wr = wb + (size_t)(h * L_ + l) * (D_ / 4);
        float acc = 0.f;
        #pragma unroll 4
        for (int k = 0; k < D_ / 4; ++k) {
            float4 a = xr[k], b = wr[k];
            acc += a.x * b.x + a.y * b.y + a.z * b.z + a.w * b.w;
        }
        acc *= INV_TEMP_;
        if (mask[n * L_ + l] == 0) acc = -1e9f;
        logit[h][l] = acc;
    }
    __syncthreads();

    // softmax: wave wv owns row h=wv (wave32: 32 lanes x 4 elems)
    const int wv = tid >> 5, lane = tid & 31;
    float v[4];
    float m = -INFINITY;
    #pragma unroll
    for (int j = 0; j < 4; ++j) { v[j] = logit[wv][lane + 32 * j]; m = fmaxf(m, v[j]); }
    #pragma unroll
    for (int off = 16; off >= 1; off >>= 1) m = fmaxf(m, __shfl_xor(m, off, 32));
    float s = 0.f;
    #pragma unroll
    for (int j = 0; j < 4; ++j) { v[j] = __expf(v[j] - m); s += v[j]; }
    #pragma unroll
    for (int off = 16; off >= 1; off >>= 1) s += __shfl_xor(s, off, 32);
    const float inv = 1.f / s;
    float* dst = attn_out + ((size_t)n * H_ + wv) * L_;
    #pragma unroll
    for (int j = 0; j < 4; ++j) dst[lane + 32 * j] = v[j] * inv;
}

// ---------------------------------------------------------------------------
// Kernel 2: v = x @ v_w^T   (M=16384, N=512, K=512), f32 WMMA 16x16x4.
// Block = 8 waves, 16 rows x 512 cols; each wave owns a 16x64 strip (4 N
// tiles) so every A fragment is reused by 4 WMMAs. A tile staged via
// async global->LDS B128.
// ---------------------------------------------------------------------------
__global__ __launch_bounds__(256) void vproj_kernel(
    const float* __restrict__ x, const float* __restrict__ vw,
    float* __restrict__ vout)
{
    __shared__ float a_tile[16][D_];   // 32 KB
    const int tid  = threadIdx.x;
    const int row0 = blockIdx.x * 16;

    for (int i = tid; i < 16 * (D_ / 4); i += 256) {
        int r = i >> 7, c4 = i & 127;
        async_load_b128(x + (size_t)(row0 + r) * D_ + c4 * 4, &a_tile[r][c4 * 4]);
    }
    wait_async_lds();
    __syncthreads();

    const int wv = tid >> 5, lane = tid & 31;
    const int mr = lane & 15, hi = lane >> 4;   // A: row mr; B: col mr
    const int cn0 = wv * 64;
    const float* brow[4];
    #pragma unroll
    for (int t = 0; t < 4; ++t) brow[t] = vw + (size_t)(cn0 + t * 16 + mr) * D_;

    v8f acc[4] = {};
    #pragma unroll 4
    for (int i = 0; i < D_ / 4; ++i) {
        int k = 4 * i + 2 * hi;
        v2f a = { a_tile[mr][k], a_tile[mr][k + 1] };   // reused 4x
        #pragma unroll
        for (int t = 0; t < 4; ++t) {
            v2f b = { brow[t][k], brow[t][k + 1] };
            acc[t] = __builtin_amdgcn_wmma_f32_16x16x4_f32(false, a, false, b,
                                                           (short)0, acc[t], false, false);
        }
    }
    #pragma unroll
    for (int t = 0; t < 4; ++t) {
        float* out = vout + (size_t)row0 * D_ + cn0 + t * 16;
        #pragma unroll
        for (int r = 0; r < 8; ++r)
            out[(size_t)(r + 8 * hi) * D_ + mr] = acc[t][r];   // C: vgpr r -> M=r(+8)
    }
}

// ---------------------------------------------------------------------------
// Kernel 3: out[h,d] = sum_l attn[h,l] * v[l,d] per token, then LayerNorm(d).
// One block per token; A = attn padded 8->16 rows; K=128. A fragment hoisted
// over the 4 N tiles of each wave.
// ---------------------------------------------------------------------------
__global__ __launch_bounds__(256) void readout_ln_kernel(
    const float* __restrict__ attn, const float* __restrict__ v,
    const float* __restrict__ g, const float* __restrict__ bta,
    float* __restrict__ out_ln)
{
    __shared__ float a_t[16][L_];   // 8 KB, rows 8..15 zero-padded
    __shared__ float o_t[16][D_];   // 32 KB
    const int n = blockIdx.x, tid = threadIdx.x;

    for (int i = tid; i < 16 * L_; i += 256) {
        int r = i >> 7, l = i & 127;
        a_t[r][l] = (r < H_) ? attn[((size_t)n * H_ + r) * L_ + l] : 0.f;
    }
    __syncthreads();

    const int wv = tid >> 5, lane = tid & 31;
    const int mr = lane & 15, hi = lane >> 4;
    const int cn0 = wv * 64;
    const float* vb = v + (size_t)n * L_ * D_;

    v8f acc[4] = {};
    #pragma unroll 4
    for (int i = 0; i < L_ / 4; ++i) {
        int k = 4 * i + 2 * hi;
        v2f a = { a_t[mr][k], a_t[mr][k + 1] };
        const float* vk0 = vb + (size_t)k * D_ + mr;
        #pragma unroll
        for (int t = 0; t < 4; ++t) {
            v2f b = { vk0[cn0 + t * 16], vk0[D_ + cn0 + t * 16] };
            acc[t] = __builtin_amdgcn_wmma_f32_16x16x4_f32(false, a, false, b,
                                                           (short)0, acc[t], false, false);
        }
    }
    #pragma unroll
    for (int t = 0; t < 4; ++t)
        #pragma unroll
        for (int r = 0; r < 8; ++r)
            o_t[r + 8 * hi][cn0 + t * 16 + mr] = acc[t][r];
    __syncthreads();

    // LayerNorm: wave wv owns head row wv (rows 8..15 are padding)
    float s = 0.f, sq = 0.f, vals[16];
    #pragma unroll
    for (int j = 0; j < 16; ++j) {
        float xv = o_t[wv][lane + 32 * j];
        vals[j] = xv; s += xv; sq += xv * xv;
    }
    #pragma unroll
    for (int off = 16; off >= 1; off >>= 1) {
        s  += __shfl_xor(s,  off, 32);
        sq += __shfl_xor(sq, off, 32);
    }
    const float mu   = s  * (1.f / D_);
    const float var  = sq * (1.f / D_) - mu * mu;
    const float rstd = rsqrtf(var + EPS_);
    float* dst = out_ln + ((size_t)n * H_ + wv) * D_;
    #pragma unroll
    for (int j = 0; j < 16; ++j) {
        int d = lane + 32 * j;
        dst[d] = (vals[j] - mu) * rstd * g[d] + bta[d];
    }
}

// ---------------------------------------------------------------------------
// Kernel 4: h = relu(out_ln @ w1^T + b1)  (M=1024, N=2048, K=512)
// Block = 16 rows x 512 cols (wave = 16x64 strip); grid.y covers N/512.
// ---------------------------------------------------------------------------
__global__ __launch_bounds__(256) void ffn1_kernel(
    const float* __restrict__ a, const float* __restrict__ w1,
    const float* __restrict__ b1, float* __restrict__ hbuf)
{
    __shared__ float a_tile[16][D_];
    const int tid  = threadIdx.x;
    const int row0 = blockIdx.x * 16;
    const int colB = blockIdx.y * 512;

    for (int i = tid; i < 16 * (D_ / 4); i += 256) {
        int r = i >> 7, c4 = i & 127;
        async_load_b128(a + (size_t)(row0 + r) * D_ + c4 * 4, &a_tile[r][c4 * 4]);
    }
    wait_async_lds();
    __syncthreads();

    const int wv = tid >> 5, lane = tid & 31;
    const int mr = lane & 15, hi = lane >> 4;
    const int cn0 = colB + wv * 64;
    const float* brow[4];
    #pragma unroll
    for (int t = 0; t < 4; ++t) brow[t] = w1 + (size_t)(cn0 + t * 16 + mr) * D_;

    v8f acc[4] = {};
    #pragma unroll 4
    for (int i = 0; i < D_ / 4; ++i) {
        int k = 4 * i + 2 * hi;
        v2f av = { a_tile[mr][k], a_tile[mr][k + 1] };
        #pragma unroll
        for (int t = 0; t < 4; ++t) {
            v2f bv = { brow[t][k], brow[t][k + 1] };
            acc[t] = __builtin_amdgcn_wmma_f32_16x16x4_f32(false, av, false, bv,
                                                           (short)0, acc[t], false, false);
        }
    }
    #pragma unroll
    for (int t = 0; t < 4; ++t) {
        const float bias = b1[cn0 + t * 16 + mr];
        float* out = hbuf + (size_t)row0 * DI_ + cn0 + t * 16;
        #pragma unroll
        for (int r = 0; r < 8; ++r)
            out[(size_t)(r + 8 * hi) * DI_ + mr] = fmaxf(acc[t][r] + bias, 0.f);
    }
}

// ---------------------------------------------------------------------------
// Kernel 5: y = LN(h @ w2^T + b2 + out_ln)  (M=1024, N=512, K=2048)
// Block owns 16 full rows (16x512) so the final LN is done in-block.
// A staged in 4 K-chunks of 512 via async B128 loads through one 32 KB LDS
// buffer, reused for the output tile in the epilogue.
// ---------------------------------------------------------------------------
__global__ __launch_bounds__(256) void ffn2_ln_kernel(
    const float* __restrict__ hbuf, const float* __restrict__ w2,
    const float* __restrict__ b2, const float* __restrict__ resid,
    const float* __restrict__ g, const float* __restrict__ bta,
    float* __restrict__ y)
{
    __shared__ float smem[16][D_];   // A chunk, then output tile
    const int tid  = threadIdx.x;
    const int row0 = blockIdx.x * 16;
    const int wv = tid >> 5, lane = tid & 31;
    const int mr = lane & 15, hi = lane >> 4;
    const int cn0 = wv * 64;

    v8f acc[4] = {};
    for (int kc = 0; kc < DI_ / D_; ++kc) {
        __syncthreads();     // all waves done reading previous chunk
        for (int i = tid; i < 16 * (D_ / 4); i += 256) {
            int r = i >> 7, c4 = i & 127;
            async_load_b128(hbuf + (size_t)(row0 + r) * DI_ + kc * D_ + c4 * 4,
                            &smem[r][c4 * 4]);
        }
        wait_async_lds();
        __syncthreads();

        #pragma unroll 4
        for (int i = 0; i < D_ / 4; ++i) {
            int k = 4 * i + 2 * hi;
            v2f av = { smem[mr][k], smem[mr][k + 1] };
            #pragma unroll
            for (int t = 0; t < 4; ++t) {
                const float* brow = w2 + (size_t)(cn0 + t * 16 + mr) * DI_ + kc * D_;
                v2f bv = { brow[k], brow[k + 1] };
                acc[t] = __builtin_amdgcn_wmma_f32_16x16x4_f32(false, av, false, bv,
                                                               (short)0, acc[t], false, false);
            }
        }
    }
    __syncthreads();

    // epilogue: bias + residual into smem
    #pragma unroll
    for (int t = 0; t < 4; ++t) {
        const int col = cn0 + t * 16 + mr;
        const float bias = b2[col];
        #pragma unroll
        for (int r = 0; r < 8; ++r) {
            int row = r + 8 * hi;
            smem[row][col] = acc[t][r] + bias + resid[(size_t)(row0 + row) * D_ + col];
        }
    }
    __syncthreads();

    // final LayerNorm: 16 rows / 8 waves = 2 rows each
    #pragma unroll
    for (int rr = 0; rr < 2; ++rr) {
        const int row = wv * 2 + rr;
        float s = 0.f, sq = 0.f, vals[16];
        #pragma unroll
        for (int j = 0; j < 16; ++j) {
            float xv = smem[row][lane + 32 * j];
            vals[j] = xv; s += xv; sq += xv * xv;
        }
        #pragma unroll
        for (int off = 16; off >= 1; off >>= 1) {
            s  += __shfl_xor(s,  off, 32);
            sq += __shfl_xor(sq, off, 32);
        }
        const float mu   = s  * (1.f / D_);
        const float var  = sq * (1.f / D_) - mu * mu;
        const float rstd = rsqrtf(var + EPS_);
        float* dst = y + (size_t)(row0 + row) * D_;
        #pragma unroll
        for (int j = 0; j < 16; ++j) {
            int d = lane + 32 * j;
            dst[d] = (vals[j] - mu) * rstd * g[d] + bta[d];
        }
    }
}

// ---------------------------------------------------------------------------
extern "C" void kernel_launch(void* const* d_in, const int* in_sizes, int n_in,
                              void* d_out, int out_size, void* d_ws, size_t ws_size,
                              hipStream_t stream) {
    (void)in_sizes; (void)n_in; (void)out_size; (void)ws_size;
    const float* x    = (const float*)d_in[0];
    const int*   mask = (const int*)  d_in[1];
    const float* w    = (const float*)d_in[2];
    const float* v_w  = (const float*)d_in[3];
    const float* ln_g = (const float*)d_in[4];
    const float* ln_b = (const float*)d_in[5];
    const float* w1   = (const float*)d_in[6];
    const float* b1   = (const float*)d_in[7];
    const float* w2   = (const float*)d_in[8];
    const float* b2   = (const float*)d_in[9];
    const float* fg   = (const float*)d_in[10];
    const float* fb   = (const float*)d_in[11];

    float* y_out    = (float*)d_out;                    // (B,S,H,D) = 524288 f32
    float* attn_out = y_out + (size_t)NH_ * D_;         // (B,S,H,L) = 131072 f32

    float* ws     = (float*)d_ws;
    float* v_ws   = ws;                                 // (N,L,D)  32 MB
    float* out_ln = v_ws   + (size_t)N_ * L_ * D_;      // (NH,D)    2 MB
    float* hbuf   = out_ln + (size_t)NH_ * D_;          // (NH,DI)   8 MB

    attn_softmax_kernel<<<N_, 256, 0, stream>>>(x, mask, w, attn_out);
    vproj_kernel<<<N_ * L_ / 16, 256, 0, stream>>>(x, v_w, v_ws);
    readout_ln_kernel<<<N_, 256, 0, stream>>>(attn_out, v_ws, ln_g, ln_b, out_ln);
    ffn1_kernel<<<dim3(NH_ / 16, DI_ / 512), 256, 0, stream>>>(out_ln, w1, b1, hbuf);
    ffn2_ln_kernel<<<NH_ / 16, 256, 0, stream>>>(hbuf, w2, b2, out_ln, fg, fb, y_out);
}